// GATGNN_19937238188413
// MI455X (gfx1250) — compile-verified
//
#include <hip/hip_runtime.h>

typedef float v2f __attribute__((ext_vector_type(2)));
typedef float v8f __attribute__((ext_vector_type(8)));

#define DIN 128
#define C1V 16
#define C2V 32
#define NEG_SLOPE 0.2f

// ---------- order-preserving float <-> uint key (for atomic segment-max) ----
__device__ __forceinline__ unsigned fkey(float f) {
  unsigned u = __float_as_uint(f);
  return (u & 0x80000000u) ? ~u : (u | 0x80000000u);
}
__device__ __forceinline__ float fval(unsigned k) {
  return __uint_as_float((k & 0x80000000u) ? (k ^ 0x80000000u) : ~k);
}
__device__ __forceinline__ float lrelu(float v) {
  return v > 0.0f ? v : NEG_SLOPE * v;
}

// ---------------------------------------------------------------------------
// GEMM layer 1: xl = x@Wl + bl ; xr = x@Wr + br    (K=128 -> Cout=16)
// fp32 WMMA 16x16x4, one 16-row tile per wave, both outputs share the A tile.
// A (16x4 f32): lanes 0-15 hold K=k,k+1 ; lanes 16-31 hold K=k+2,k+3
// B (4x16 f32): mirrored striping; C/D: VGPR v -> M = v + 8*(lane>=16)
// ---------------------------------------------------------------------------
__global__ void gat_gemm1(const float* __restrict__ x,
                          const float* __restrict__ Wl, const float* __restrict__ bl,
                          const float* __restrict__ Wr, const float* __restrict__ br,
                          float* __restrict__ xl, float* __restrict__ xr, int nTiles) {
  int wave = threadIdx.x >> 5;
  int lane = threadIdx.x & 31;
  int tile = blockIdx.x * (blockDim.x >> 5) + wave;
  if (tile >= nTiles) return;                  // wave-uniform: EXEC stays all-1s
  int hi = lane >> 4;                          // lane-group select (K pair)
  int ml = lane & 15;                          // A row (M) / B,C column (N)
  const float* arow = x + (size_t)(tile * 16 + ml) * DIN + 2 * hi;
  v8f accL = {}; v8f accR = {};
  #pragma unroll 4
  for (int k = 0; k < DIN; k += 4) {
    v2f a; a.x = arow[k]; a.y = arow[k + 1];
    int r0 = (k + 2 * hi) * C1V + ml, r1 = r0 + C1V;
    v2f bL; bL.x = Wl[r0]; bL.y = Wl[r1];
    v2f bR; bR.x = Wr[r0]; bR.y = Wr[r1];
    accL = __builtin_amdgcn_wmma_f32_16x16x4_f32(false, a, false, bL, (short)0, accL, false, false);
    accR = __builtin_amdgcn_wmma_f32_16x16x4_f32(false, a, false, bR, (short)0, accR, false, false);
  }
  float cl = bl[ml], cr = br[ml];
  #pragma unroll
  for (int v = 0; v < 8; ++v) {
    size_t row = (size_t)tile * 16 + v + 8 * hi;
    xl[row * C1V + ml] = accL[v] + cl;
    xr[row * C1V + ml] = accR[v] + cr;
  }
}

// ---------------------------------------------------------------------------
// GEMM layer 2: xl2 = h@Wl2 + bl2 ; xr2 = h@Wr2 + br2  (K=16 -> Cout=32)
// Two 16-col output tiles per matrix -> 4 f32 accumulators per wave.
// ---------------------------------------------------------------------------
__global__ void gat_gemm2(const float* __restrict__ h,
                          const float* __restrict__ Wl, const float* __restrict__ bl,
                          const float* __restrict__ Wr, const float* __restrict__ br,
                          float* __restrict__ xl, float* __restrict__ xr, int nTiles) {
  int wave = threadIdx.x >> 5;
  int lane = threadIdx.x & 31;
  int tile = blockIdx.x * (blockDim.x >> 5) + wave;
  if (tile >= nTiles) return;
  int hi = lane >> 4;
  int ml = lane & 15;
  const float* arow = h + (size_t)(tile * 16 + ml) * C1V + 2 * hi;
  v8f aL0 = {}, aL1 = {}, aR0 = {}, aR1 = {};
  #pragma unroll
  for (int k = 0; k < C1V; k += 4) {
    v2f a; a.x = arow[k]; a.y = arow[k + 1];
    int r0 = (k + 2 * hi) * C2V + ml, r1 = r0 + C2V;
    v2f b;
    b.x = Wl[r0];      b.y = Wl[r1];
    aL0 = __builtin_amdgcn_wmma_f32_16x16x4_f32(false, a, false, b, (short)0, aL0, false, false);
    b.x = Wl[r0 + 16]; b.y = Wl[r1 + 16];
    aL1 = __builtin_amdgcn_wmma_f32_16x16x4_f32(false, a, false, b, (short)0, aL1, false, false);
    b.x = Wr[r0];      b.y = Wr[r1];
    aR0 = __builtin_amdgcn_wmma_f32_16x16x4_f32(false, a, false, b, (short)0, aR0, false, false);
    b.x = Wr[r0 + 16]; b.y = Wr[r1 + 16];
    aR1 = __builtin_amdgcn_wmma_f32_16x16x4_f32(false, a, false, b, (short)0, aR1, false, false);
  }
  float cl0 = bl[ml], cl1 = bl[ml + 16], cr0 = br[ml], cr1 = br[ml + 16];
  #pragma unroll
  for (int v = 0; v < 8; ++v) {
    size_t row = (size_t)tile * 16 + v + 8 * hi;
    xl[row * C2V + ml]      = aL0[v] + cl0;
    xl[row * C2V + ml + 16] = aL1[v] + cl1;
    xr[row * C2V + ml]      = aR0[v] + cr0;
    xr[row * C2V + ml + 16] = aR1[v] + cr1;
  }
}

// ---------------------------------------------------------------------------
// Init: zero the aggregation buffer, reset segment max-keys / sums
// ---------------------------------------------------------------------------
__global__ void gat_init1(float* __restrict__ agg, unsigned* __restrict__ mkey,
                          float* __restrict__ ssum, int n16, int n) {
  int i = blockIdx.x * blockDim.x + threadIdx.x;
  if (i < n16) agg[i] = 0.0f;
  if (i < n) { mkey[i] = 0u; ssum[i] = 0.0f; }
}
__global__ void gat_init2(unsigned* __restrict__ mkey, float* __restrict__ ssum, int n) {
  int i = blockIdx.x * blockDim.x + threadIdx.x;
  if (i < n) { mkey[i] = 0u; ssum[i] = 0.0f; }
}

// ---------------------------------------------------------------------------
// Edge scores: e = sum_c att[c] * lrelu(xl[src][c] + xr[dst][c] + w*We[c])
// plus atomic segment-max over dst (order-preserving uint keys).
// C = 16 (layer 1, w = edge_weight) or 32 (layer 2, w = alpha1; ew aliases eout)
// ---------------------------------------------------------------------------
template <int C>
__global__ void gat_edge_score(const int* __restrict__ src, const int* __restrict__ dst,
                               const float* __restrict__ ew,
                               const float* __restrict__ xl, const float* __restrict__ xr,
                               const float* __restrict__ We, const float* __restrict__ att,
                               float* __restrict__ eout, unsigned* __restrict__ mkey, int E) {
  int e = blockIdx.x * blockDim.x + threadIdx.x;
  if (e >= E) return;
  int s = src[e], d = dst[e];
  float w = ew[e];
  const float* pl = xl + (size_t)s * C;
  const float* pr = xr + (size_t)d * C;
  float sc = 0.0f;
  #pragma unroll
  for (int c = 0; c < C; ++c)
    sc += att[c] * lrelu(pl[c] + pr[c] + w * We[c]);
  eout[e] = sc;
  atomicMax(&mkey[d], fkey(sc));
}

// exp(e - max[dst]) -> eout, atomic segment-sum of exps
__global__ void gat_edge_exp(const int* __restrict__ dst, float* __restrict__ ebuf,
                             const unsigned* __restrict__ mkey, float* __restrict__ ssum, int E) {
  int e = blockIdx.x * blockDim.x + threadIdx.x;
  if (e >= E) return;
  int d = dst[e];
  float ex = __expf(ebuf[e] - fval(mkey[d]));
  ebuf[e] = ex;
  atomicAdd(&ssum[d], ex);
}

// alpha = ex / (sum[dst]+eps) -> ebuf ; scatter-add alpha * xl1[src] into agg
__global__ void gat_edge_aggregate(const int* __restrict__ src, const int* __restrict__ dst,
                                   float* __restrict__ ebuf, const float* __restrict__ ssum,
                                   const float* __restrict__ xl, float* __restrict__ agg, int E) {
  int e = blockIdx.x * blockDim.x + threadIdx.x;
  if (e >= E) return;
  int s = src[e], d = dst[e];
  float alpha = ebuf[e] / (ssum[d] + 1e-16f);
  ebuf[e] = alpha;                          // alpha1 becomes layer-2 edge feature
  const float* pl = xl + (size_t)s * C1V;
  float* pa = agg + (size_t)d * C1V;
  #pragma unroll
  for (int c = 0; c < C1V; ++c)
    atomicAdd(&pa[c], alpha * pl[c]);
}

// h = relu(agg + b1)  (H=1 so mean over heads is identity)
__global__ void gat_finalize_h(float* __restrict__ h, const float* __restrict__ b1, int n16) {
  int i = blockIdx.x * blockDim.x + threadIdx.x;
  if (i >= n16) return;
  float v = h[i] + b1[i & (C1V - 1)];
  h[i] = v > 0.0f ? v : 0.0f;
}

// alpha2 = ex / (sum[dst]+eps) -> out
__global__ void gat_write_alpha2(const int* __restrict__ dst, const float* __restrict__ ebuf,
                                 const float* __restrict__ ssum, float* __restrict__ out, int E) {
  int e = blockIdx.x * blockDim.x + threadIdx.x;
  if (e >= E) return;
  out[e] = ebuf[e] / (ssum[dst[e]] + 1e-16f);
}

__global__ void gat_copy_edges(const int* __restrict__ edges, int* __restrict__ out, int n) {
  int i = blockIdx.x * blockDim.x + threadIdx.x;
  if (i < n) out[i] = edges[i];
}

// ---------------------------------------------------------------------------
extern "C" void kernel_launch(void* const* d_in, const int* in_sizes, int n_in,
                              void* d_out, int out_size, void* d_ws, size_t ws_size,
                              hipStream_t stream) {
  const float* x    = (const float*)d_in[0];
  const int*   edges= (const int*)  d_in[1];
  const float* ew   = (const float*)d_in[2];
  const float* Wl1  = (const float*)d_in[3];
  const float* bl1  = (const float*)d_in[4];
  const float* Wr1  = (const float*)d_in[5];
  const float* br1  = (const float*)d_in[6];
  const float* We1  = (const float*)d_in[7];
  const float* att1 = (const float*)d_in[8];
  const float* b1   = (const float*)d_in[9];
  const float* Wl2  = (const float*)d_in[10];
  const float* bl2  = (const float*)d_in[11];
  const float* Wr2  = (const float*)d_in[12];
  const float* br2  = (const float*)d_in[13];
  const float* We2  = (const float*)d_in[14];
  const float* att2 = (const float*)d_in[15];

  const int N = in_sizes[0] / DIN;
  const int E = in_sizes[1] / 2;
  const int* src = edges;
  const int* dst = edges + E;

  // workspace carve-out (floats)
  float*    ws   = (float*)d_ws;
  float*    xl1  = ws;                         // 16N
  float*    xr1  = xl1 + (size_t)16 * N;       // 16N
  float*    hbuf = xr1 + (size_t)16 * N;       // 16N  (agg -> h, in place)
  float*    xl2  = hbuf + (size_t)16 * N;      // 32N
  float*    xr2  = xl2 + (size_t)32 * N;       // 32N
  unsigned* mkey = (unsigned*)(xr2 + (size_t)32 * N); // N
  float*    ssum = (float*)(mkey + N);         // N
  float*    ebuf = ssum + N;                   // E  (score -> exp -> alpha1 -> score2 -> exp2)

  const int nTiles = N / 16;                   // N = 100000 is a multiple of 16
  const int gemmBlk = 128;                     // 4 waves / block, 1 tile / wave
  const int gemmGrid = (nTiles + 3) / 4;
  const int eBlk = 256;
  const int eGrid = (E + eBlk - 1) / eBlk;
  const int n16 = 16 * N;
  const int nGrid16 = (n16 + 255) / 256;
  const int nGrid = (N + 255) / 256;

  // ---- layer 1 ----
  gat_init1<<<nGrid16, 256, 0, stream>>>(hbuf, mkey, ssum, n16, N);
  gat_gemm1<<<gemmGrid, gemmBlk, 0, stream>>>(x, Wl1, bl1, Wr1, br1, xl1, xr1, nTiles);
  gat_edge_score<C1V><<<eGrid, eBlk, 0, stream>>>(src, dst, ew, xl1, xr1, We1, att1, ebuf, mkey, E);
  gat_edge_exp<<<eGrid, eBlk, 0, stream>>>(dst, ebuf, mkey, ssum, E);
  gat_edge_aggregate<<<eGrid, eBlk, 0, stream>>>(src, dst, ebuf, ssum, xl1, hbuf, E);
  gat_finalize_h<<<nGrid16, 256, 0, stream>>>(hbuf, b1, n16);

  // ---- layer 2 (only alpha2 needed; skip final aggregation) ----
  gat_init2<<<nGrid, 256, 0, stream>>>(mkey, ssum, N);
  gat_gemm2<<<gemmGrid, gemmBlk, 0, stream>>>(hbuf, Wl2, bl2, Wr2, br2, xl2, xr2, nTiles);
  gat_edge_score<C2V><<<eGrid, eBlk, 0, stream>>>(src, dst, ebuf, xl2, xr2, We2, att2, ebuf, mkey, E);
  gat_edge_exp<<<eGrid, eBlk, 0, stream>>>(dst, ebuf, mkey, ssum, E);

  // ---- outputs: edges (2E ints) then alpha2 (E floats) ----
  gat_write_alpha2<<<eGrid, eBlk, 0, stream>>>(dst, ebuf, ssum, (float*)d_out + (size_t)2 * E, E);
  gat_copy_edges<<<(2 * E + 255) / 256, 256, 0, stream>>>(edges, (int*)d_out, 2 * E);
}